// SVConv2d_62989990363670
// MI455X (gfx1250) — compile-verified
//
#include <hip/hip_runtime.h>
#include <stdint.h>

typedef __attribute__((ext_vector_type(2))) float v2f;
typedef __attribute__((ext_vector_type(8))) float v8f;

#define N_    8
#define CIN   32
#define COUT  32
#define CP    288      // Cin * K * K
#define H_    64
#define W_    64
#define HW    4096
#define CPB   16       // K-chunk per LDS stage
#define PXP   20       // padded pixel stride (conflict-free, 16B-alignable)

// LDS: weight chunk [cpi][o][px], patches chunk [cpi][n(16)][px]
// 16*32*20*4 = 40 KB  +  16*16*20*4 = 20 KB  -> 60 KB per workgroup

__global__ __launch_bounds__(512)
void svconv_wmma_f32(const float* __restrict__ x,
                     const float* __restrict__ w,
                     const float* __restrict__ bias,
                     float* __restrict__ out) {
    __shared__ float lds_w[CPB * COUT * PXP];
    __shared__ float lds_x[CPB * 16 * PXP];

    const int tid  = threadIdx.x;
    const int lane = tid & 31;
    const int wave = tid >> 5;          // pixel index within the 16-pixel tile
    const int bid  = blockIdx.x;
    const int h    = bid >> 2;          // row
    const int w0   = (bid & 3) << 4;    // first pixel of tile

    // Zero-fill lds_x once: columns n=8..15 stay zero forever (padding the
    // 8-wide batch up to the 16 WMMA columns); n<8 rewritten every chunk.
    for (int i = tid; i < CPB * 16 * PXP; i += 512) lds_x[i] = 0.0f;

    v8f acc0 = {};   // Cout 0..15
    v8f acc1 = {};   // Cout 16..31

    const int m   = lane & 15;          // A row / B col within tile
    const int kh2 = (lane >> 4) << 1;   // K-row base for this half-wave (0 or 2)
    const int px  = wave;

    for (int cp0 = 0; cp0 < CP; cp0 += CPB) {
        __syncthreads();   // WAR: previous chunk's compute done before overwrite

        // ---- stage weight chunk: 32 Cout x 16 cp x 16 px ----
        // Direct global -> LDS async DMA (no VGPR staging), b128 over px.
        // 8192 floats = 2048 b128 transfers; 512 threads -> 4 iterations.
        #pragma unroll
        for (int it = 0; it < 4; ++it) {
            int l   = tid + it * 512;        // 0..2047
            int px4 = (l & 3) << 2;          // 0,4,8,12
            int cpi = (l >> 2) & 15;
            int o   = l >> 6;                // 0..31
            size_t   gi    = (size_t)(o * CP + cp0 + cpi) * HW + (size_t)(h * W_ + w0 + px4);
            uint64_t gaddr = (uint64_t)(uintptr_t)(w + gi);
            unsigned laddr = (unsigned)(uintptr_t)&lds_w[(cpi * COUT + o) * PXP + px4];
            asm volatile("global_load_async_to_lds_b128 %0, %1, off"
                         :: "v"(laddr), "v"(gaddr)
                         : "memory");
        }

        // ---- stage im2col patches: 8 n x 16 cp x 16 px (scalar, coalesced px) ----
        // (runs while the async engine streams the weight chunk into LDS)
        #pragma unroll
        for (int it = 0; it < 4; ++it) {
            int l   = tid + it * 512;        // 0..2047
            int pxl = l & 15;
            int cpi = (l >> 4) & 15;
            int n   = l >> 8;                // 0..7
            int cp  = cp0 + cpi;
            int c   = cp / 9;
            int p   = cp - c * 9;
            int kh  = p / 3;
            int kw  = p - kh * 3;
            int hh  = h + kh - 1;
            int ww  = w0 + pxl + kw - 1;
            float v = 0.0f;
            if ((unsigned)hh < (unsigned)H_ && (unsigned)ww < (unsigned)W_)
                v = x[((n * CIN + c) * H_ + hh) * W_ + ww];
            lds_x[(cpi * 16 + n) * PXP + pxl] = v;
        }

        // All of this wave's async LDS fills complete, then publish to WG.
        asm volatile("s_wait_asynccnt 0x0" ::: "memory");
        __syncthreads();   // RAW: LDS populated before compute

        // ---- per-wave 32x8x16 GEMM step on its pixel ----
        #pragma unroll
        for (int s = 0; s < 4; ++s) {
            int kr = (s << 2) + kh2;         // this half-wave's K rows: kr, kr+1
            v2f a0, a1, b;
            // A (weight) 16x4: lane m = Cout row, VGPR0/1 = K rows kr/kr+1
            a0.x = lds_w[( kr      * COUT +       m) * PXP + px];
            a0.y = lds_w[((kr + 1) * COUT +       m) * PXP + px];
            a1.x = lds_w[( kr      * COUT + 16 +  m) * PXP + px];
            a1.y = lds_w[((kr + 1) * COUT + 16 +  m) * PXP + px];
            // B (patches) 4x16: lane m = batch column, rows kr/kr+1
            b.x  = lds_x[( kr      * 16 + m) * PXP + px];
            b.y  = lds_x[((kr + 1) * 16 + m) * PXP + px];
            acc0 = __builtin_amdgcn_wmma_f32_16x16x4_f32(
                       false, a0, false, b, (short)0, acc0, false, false);
            acc1 = __builtin_amdgcn_wmma_f32_16x16x4_f32(
                       false, a1, false, b, (short)0, acc1, false, false);
        }
    }

    // ---- epilogue: D layout VGPR r = row (r + 8*half), col = lane&15 ----
    const int n     = lane & 15;         // batch column
    const int mhalf = (lane >> 4) << 3;  // 0 or 8
    if (n < N_) {
        #pragma unroll
        for (int r = 0; r < 8; ++r) {
            int o0 = r + mhalf;          // Cout 0..15
            int o1 = 16 + r + mhalf;     // Cout 16..31
            size_t oi0 = (size_t)(n * COUT + o0) * HW + (size_t)(h * W_ + w0 + px);
            size_t oi1 = (size_t)(n * COUT + o1) * HW + (size_t)(h * W_ + w0 + px);
            out[oi0] = acc0[r] + bias[o0];
            out[oi1] = acc1[r] + bias[o1];
        }
    }
}

extern "C" void kernel_launch(void* const* d_in, const int* in_sizes, int n_in,
                              void* d_out, int out_size, void* d_ws, size_t ws_size,
                              hipStream_t stream) {
    (void)in_sizes; (void)n_in; (void)d_ws; (void)ws_size; (void)out_size;
    const float* x    = (const float*)d_in[0];
    const float* wgt  = (const float*)d_in[1];
    const float* bias = (const float*)d_in[2];
    float* out        = (float*)d_out;
    // 64 rows * 4 pixel-tiles = 256 blocks; 512 threads = 16 wave32 (1 pixel/wave)
    svconv_wmma_f32<<<dim3(H_ * (W_ / 16)), dim3(512), 0, stream>>>(x, wgt, bias, out);
}